// DualChannelRec_13477607374930
// MI455X (gfx1250) — compile-verified
//
#include <hip/hip_runtime.h>
#include <stdint.h>

// ---------------------------------------------------------------------------
// DualChannelRec: 3x COO SpMM (N=150000 nodes, 64 feats, NNZ=4M) + telescoped
// low/high-pass filters.  L2-resident dense state, HBM only streams edges.
// CDNA5 paths used: global_load_async_to_lds_b32 (ASYNCcnt, double-buffered
// edge staging) + s_wait_asynccnt, native global_atomic_add_f32 scatter.
// ---------------------------------------------------------------------------

#define USERS    100000
#define ITEMS    50000
#define NNODES   (USERS + ITEMS)          // 150000
#define LATDIM   64
#define NNZ_     4000000
#define BATCH_   4096

#define CHUNK    256                      // edges staged per LDS buffer
#define CPB      4                        // chunks per block (pipeline depth 2)
#define EPB      (CHUNK * CPB)            // 1024 edges per block

// low 32 bits of a generic pointer into LDS == byte offset within the
// workgroup LDS allocation (shared aperture base has zero low bits).
__device__ __forceinline__ uint32_t lds_off(const void* p) {
  return (uint32_t)(uintptr_t)p;
}

// async DMA of one dword per lane: global -> LDS (tracked by ASYNCcnt)
__device__ __forceinline__ void async_ld_b32(uint32_t lds, const void* gptr) {
  asm volatile("global_load_async_to_lds_b32 %0, %1, off"
               :: "v"(lds), "v"(gptr) : "memory");
}

__device__ __forceinline__ void atomic_add_f32(float* p, float v) {
  (void)__hip_atomic_fetch_add(p, v, __ATOMIC_RELAXED, __HIP_MEMORY_SCOPE_AGENT);
}

// ---------------------------------------------------------------------------
// y[row] += val * x[col]   (y must be pre-zeroed).  256 threads/block.
// 16 lanes per edge, float4 per lane -> 256B coalesced gather per edge.
// Edge triplets are staged into LDS with a double-buffered async pipeline.
// ---------------------------------------------------------------------------
__global__ void spmm_scatter(const int* __restrict__ rows,
                             const int* __restrict__ cols,
                             const float* __restrict__ vals,
                             const float* __restrict__ x,
                             float* __restrict__ y) {
  __shared__ int   s_rows[2][CHUNK];
  __shared__ int   s_cols[2][CHUNK];
  __shared__ float s_vals[2][CHUNK];

  const int tid = threadIdx.x;
  const long long block_base = (long long)blockIdx.x * EPB;

  // prologue: issue chunk 0 (clamp index so every wave issues the same count
  // of async loads -> uniform per-wave ASYNCcnt bookkeeping)
  {
    long long idx = block_base + tid;
    if (idx >= NNZ_) idx = NNZ_ - 1;
    async_ld_b32(lds_off(&s_rows[0][tid]), rows + idx);
    async_ld_b32(lds_off(&s_cols[0][tid]), cols + idx);
    async_ld_b32(lds_off(&s_vals[0][tid]), vals + idx);
  }

  const int lane_e = tid >> 4;          // 0..15 : edge slot within pass
  const int f      = (tid & 15) << 2;   // feature base 0,4,...,60

  for (int c = 0; c < CPB; ++c) {
    const int buf = c & 1;
    if (c + 1 < CPB) {
      long long idx = block_base + (long long)(c + 1) * CHUNK + tid;
      if (idx >= NNZ_) idx = NNZ_ - 1;
      async_ld_b32(lds_off(&s_rows[buf ^ 1][tid]), rows + idx);
      async_ld_b32(lds_off(&s_cols[buf ^ 1][tid]), cols + idx);
      async_ld_b32(lds_off(&s_vals[buf ^ 1][tid]), vals + idx);
      // 3 loads of chunk c+1 still in flight; wait for chunk c's 3 (in-order)
      asm volatile("s_wait_asynccnt 3" ::: "memory");
    } else {
      asm volatile("s_wait_asynccnt 0" ::: "memory");
    }
    __syncthreads();

    const long long chunk_base = block_base + (long long)c * CHUNK;
#pragma unroll 4
    for (int p = 0; p < 16; ++p) {
      const int le = p * 16 + lane_e;               // 0..255
      if (chunk_base + le < NNZ_) {
        const int   r   = s_rows[buf][le];
        const int   col = s_cols[buf][le];
        const float v   = s_vals[buf][le];
        const float4 xv = *(const float4*)(x + (long long)col * LATDIM + f);
        float* yp = y + (long long)r * LATDIM + f;
        atomic_add_f32(yp + 0, v * xv.x);
        atomic_add_f32(yp + 1, v * xv.y);
        atomic_add_f32(yp + 2, v * xv.z);
        atomic_add_f32(yp + 3, v * xv.w);
      }
    }
    __syncthreads();   // buffer `buf` is reused by the issue two iters ahead
  }
}

// curA = lpf = [user_emb ; item_emb], curB = 0.   (float4 index space)
__global__ void init_state(const float* __restrict__ ue,
                           const float* __restrict__ ie,
                           float* __restrict__ curA,
                           float* __restrict__ curB,
                           float* __restrict__ lpf) {
  const long long i = (long long)blockIdx.x * blockDim.x + threadIdx.x;
  const long long U4 = (long long)USERS * (LATDIM / 4);
  float4 v = (i < U4) ? ((const float4*)ue)[i] : ((const float4*)ie)[i - U4];
  ((float4*)curA)[i] = v;
  ((float4*)lpf)[i]  = v;
  ((float4*)curB)[i] = make_float4(0.f, 0.f, 0.f, 0.f);
}

// lpf += src ; optionally zero the buffer that becomes the next scatter target
__global__ void accum_zero(float* __restrict__ lpf,
                           const float* __restrict__ src,
                           float* __restrict__ zbuf,
                           int do_zero) {
  const long long i = (long long)blockIdx.x * blockDim.x + threadIdx.x;
  float4 a = ((const float4*)lpf)[i];
  float4 s = ((const float4*)src)[i];
  a.x += s.x; a.y += s.y; a.z += s.z; a.w += s.w;
  ((float4*)lpf)[i] = a;
  if (do_zero) ((float4*)zbuf)[i] = make_float4(0.f, 0.f, 0.f, 0.f);
}

// out0/out2 = lpf[ancs]; out3 = user_emb[ancs] - cur3[ancs]  (telescoped hpf)
__global__ void gather_users(const long long* __restrict__ ancs,
                             const float* __restrict__ lpf,
                             const float* __restrict__ ue,
                             const float* __restrict__ cur3,
                             float* __restrict__ out0,
                             float* __restrict__ out2,
                             float* __restrict__ out3) {
  const int i  = blockIdx.x * blockDim.x + threadIdx.x;  // 0 .. BATCH*16-1
  const int b  = i >> 4;
  const int f4 = i & 15;
  const long long u = ancs[b];
  const long long o = u * (LATDIM / 4) + f4;
  float4 l = ((const float4*)lpf)[o];
  float4 e = ((const float4*)ue)[o];
  float4 c = ((const float4*)cur3)[o];
  ((float4*)out0)[i] = l;
  ((float4*)out2)[i] = l;
  float4 h = make_float4(e.x - c.x, e.y - c.y, e.z - c.z, e.w - c.w);
  ((float4*)out3)[i] = h;
}

__global__ void copy_items(const float* __restrict__ src, float* __restrict__ dst) {
  const int i = blockIdx.x * blockDim.x + threadIdx.x;   // ITEMS*16 float4s
  ((float4*)dst)[i] = ((const float4*)src)[i];
}

// ---------------------------------------------------------------------------
extern "C" void kernel_launch(void* const* d_in, const int* in_sizes, int n_in,
                              void* d_out, int out_size, void* d_ws, size_t ws_size,
                              hipStream_t stream) {
  const int*       rows = (const int*)d_in[0];
  const int*       cols = (const int*)d_in[1];
  const float*     vals = (const float*)d_in[2];
  const long long* ancs = (const long long*)d_in[3];   // jnp.int64 per reference
  const float*     ue   = (const float*)d_in[4];
  const float*     ie   = (const float*)d_in[5];
  float* out = (float*)d_out;

  const size_t N64 = (size_t)NNODES * LATDIM;          // 9.6M floats
  float* curA = (float*)d_ws;                          // ping
  float* curB = curA + N64;                            // pong
  float* lpf  = curB + N64;                            // accumulator
  // total workspace: 3 * 38.4 MB = 115.2 MB (L2-resident on MI455X's 192 MB)

  const int nvec4   = (int)(N64 / 4);                  // 2,400,000
  const int eltBlk  = nvec4 / 256;                     // 9375 (exact)
  const int spmmBlk = (NNZ_ + EPB - 1) / EPB;          // 3907

  init_state<<<eltBlk, 256, 0, stream>>>(ue, ie, curA, curB, lpf);

  // iter 0: cur1 = A @ cur0
  spmm_scatter<<<spmmBlk, 256, 0, stream>>>(rows, cols, vals, curA, curB);
  accum_zero<<<eltBlk, 256, 0, stream>>>(lpf, curB, curA, 1);
  // iter 1: cur2 = A @ cur1
  spmm_scatter<<<spmmBlk, 256, 0, stream>>>(rows, cols, vals, curB, curA);
  accum_zero<<<eltBlk, 256, 0, stream>>>(lpf, curA, curB, 1);
  // iter 2: cur3 = A @ cur2   (cur3 ends in curB)
  spmm_scatter<<<spmmBlk, 256, 0, stream>>>(rows, cols, vals, curA, curB);
  accum_zero<<<eltBlk, 256, 0, stream>>>(lpf, curB, nullptr, 0);

  // output layout (floats): [user_batch | gnn_item | user_batch | hpf_user]
  const size_t O0 = 0;
  const size_t O1 = (size_t)BATCH_ * LATDIM;                 //   262,144
  const size_t O2 = O1 + (size_t)ITEMS * LATDIM;             // 3,462,144
  const size_t O3 = O2 + (size_t)BATCH_ * LATDIM;            // 3,724,288

  gather_users<<<(BATCH_ * 16) / 256, 256, 0, stream>>>(
      ancs, lpf, ue, curB, out + O0, out + O2, out + O3);
  copy_items<<<(ITEMS * 16) / 256, 256, 0, stream>>>(
      lpf + (size_t)USERS * LATDIM, out + O1);
}